// SelfAttention_86354612453842
// MI455X (gfx1250) — compile-verified
//
#include <hip/hip_runtime.h>
#include <hip/hip_bf16.h>
#include <math.h>

typedef float v2f __attribute__((ext_vector_type(2)));
typedef float v8f __attribute__((ext_vector_type(8)));

// D = A(16x4) * B(4x16) + C, full f32 WMMA (16x16x4), 8-arg form:
// (neg_a, A, neg_b, B, c_mod, C, reuse_a, reuse_b)
#define WMMA_F32(a, b, c) \
  __builtin_amdgcn_wmma_f32_16x16x4_f32(false, (a), false, (b), (short)0, (c), false, false)

constexpr int B_  = 4;
constexpr int C_  = 512;
constexpr int N_  = 4096;   // 64*64
constexpr int FC  = 16;     // f/g channels
constexpr int HC  = 32;     // h channels

// ---------------------------------------------------------------------------
// Stage 1: fgh = [Wf;Wg;Wh] @ x[b]  (64 x 512 @ 512 x 4096 per batch)
// Outputs: f  [B][FC][N]      (row-major, B-operand loads for QK wmma)
//          gT [B][N][FC]      (transposed, A-operand float2 loads)
//          h  [B][HC][N]      (row-major, A-operand float2 loads for PV wmma)
// Each block: one (batch, 64-token slab). x read exactly once from HBM.
// ---------------------------------------------------------------------------
__global__ __launch_bounds__(256) void fgh_kernel(
    const float* __restrict__ x,
    const float* __restrict__ Wf, const float* __restrict__ bf,
    const float* __restrict__ Wg, const float* __restrict__ bg,
    const float* __restrict__ Wh, const float* __restrict__ bh,
    float* __restrict__ f, float* __restrict__ gT, float* __restrict__ h)
{
  __shared__ float xs[64][64];    // x[cb+k][i0+i]
  __shared__ float ws[64][65];    // W[oc][cb+k], padded vs bank conflicts

  const int b   = blockIdx.x >> 6;
  const int i0  = (blockIdx.x & 63) * 64;
  const int tid = threadIdx.x;
  const int iq  = (tid & 15) << 2;   // 4 consecutive tokens
  const int ocq = (tid >> 4) << 2;   // 4 consecutive output channels

  float acc[4][4];
#pragma unroll
  for (int j = 0; j < 4; ++j) {
    int oc = ocq + j;
    float bv = (oc < 16) ? bf[oc] : (oc < 32 ? bg[oc - 16] : bh[oc - 32]);
#pragma unroll
    for (int ii = 0; ii < 4; ++ii) acc[j][ii] = bv;
  }

  const float* xb = x + (size_t)b * C_ * N_;

  for (int cb = 0; cb < C_; cb += 64) {
    __syncthreads();
#pragma unroll
    for (int t = 0; t < 16; ++t) {
      int e = tid + t * 256;
      int r = e >> 6, c = e & 63;
      xs[r][c] = xb[(size_t)(cb + r) * N_ + i0 + c];
      const float* wr = (r < 16) ? (Wf + (size_t)r * C_)
                      : (r < 32) ? (Wg + (size_t)(r - 16) * C_)
                                 : (Wh + (size_t)(r - 32) * C_);
      ws[r][c] = wr[cb + c];
    }
    __syncthreads();

    for (int k = 0; k < 64; ++k) {
      float w0 = ws[ocq + 0][k], w1 = ws[ocq + 1][k];
      float w2 = ws[ocq + 2][k], w3 = ws[ocq + 3][k];
      float xv[4];
#pragma unroll
      for (int ii = 0; ii < 4; ++ii) xv[ii] = xs[k][iq + ii];
#pragma unroll
      for (int ii = 0; ii < 4; ++ii) {
        acc[0][ii] += w0 * xv[ii];
        acc[1][ii] += w1 * xv[ii];
        acc[2][ii] += w2 * xv[ii];
        acc[3][ii] += w3 * xv[ii];
      }
    }
  }

#pragma unroll
  for (int j = 0; j < 4; ++j) {
    int oc = ocq + j;
#pragma unroll
    for (int ii = 0; ii < 4; ++ii) {
      int i = i0 + iq + ii;
      float v = acc[j][ii];
      if (oc < 16)       f[((size_t)b * FC + oc) * N_ + i] = v;
      else if (oc < 32)  gT[((size_t)b * N_ + i) * FC + (oc - 16)] = v;
      else               h[((size_t)b * HC + (oc - 32)) * N_ + i] = v;
    }
  }
}

// ---------------------------------------------------------------------------
// Stage 2: flash attention, one wave per 16-row i-block.
//   S^T[j,i] = sum_c g[c,j] f[c,i]     (A = gT tile, B = f tile, 4x WMMA K=4)
//   online softmax over j (per-lane: every VGPR of S^T shares column i)
//   O^T[c,i] += sum_j h[c,j] P^T[j,i]  (A = h tile, B = exp'd S^T tile)
// ---------------------------------------------------------------------------
__global__ __launch_bounds__(32) void attn_kernel(
    const float* __restrict__ f, const float* __restrict__ gT,
    const float* __restrict__ h, float* __restrict__ o)
{
  const int b     = blockIdx.x >> 8;
  const int iBase = (blockIdx.x & 255) * 16;
  const int lane  = threadIdx.x;
  const int li    = lane & 15;
  const int hv    = lane >> 4;     // half-wave select

  const float* fb = f  + (size_t)b * FC * N_;
  const float* gb = gT + (size_t)b * N_ * FC;
  const float* hb = h  + (size_t)b * HC * N_;

  // B-operand (f) for the QK^T wmma: loaded once per wave.
  // B[k][n]: lane holds n=li, v0 = f[kBase+2*hv][i], v1 = f[kBase+2*hv+1][i]
  v2f bF[4];
#pragma unroll
  for (int kk = 0; kk < 4; ++kk) {
    int c0 = kk * 4 + 2 * hv;
    bF[kk][0] = fb[(size_t)c0 * N_ + iBase + li];
    bF[kk][1] = fb[(size_t)(c0 + 1) * N_ + iBase + li];
  }

  v8f o0 = {};   // O^T rows c=0..15
  v8f o1 = {};   // O^T rows c=16..31
  float m = -INFINITY, l = 0.f;

  for (int jt = 0; jt < 256; ++jt) {
    const int jBase = jt * 16;

    // A-operand (gT): lane holds row j=jBase+li, float2 over K
    v2f aG[4];
#pragma unroll
    for (int kk = 0; kk < 4; ++kk)
      aG[kk] = *(const v2f*)(gb + (size_t)(jBase + li) * FC + kk * 4 + 2 * hv);

    v8f s = {};
#pragma unroll
    for (int kk = 0; kk < 4; ++kk) s = WMMA_F32(aG[kk], bF[kk], s);

    // --- online softmax (per lane == per column i) ---
    float tm = s[0];
#pragma unroll
    for (int r = 1; r < 8; ++r) tm = fmaxf(tm, s[r]);
    tm = fmaxf(tm, __shfl_xor(tm, 16, 32));   // combine j-halves
    float mNew  = fmaxf(m, tm);
    float scale = __expf(m - mNew);           // first iter: exp(-inf)=0
    v8f p;
#pragma unroll
    for (int r = 0; r < 8; ++r) p[r] = __expf(s[r] - mNew);
    float rs = 0.f;
#pragma unroll
    for (int r = 0; r < 8; ++r) rs += p[r];
    rs += __shfl_xor(rs, 16, 32);
    l = l * scale + rs;
    m = mNew;
#pragma unroll
    for (int r = 0; r < 8; ++r) { o0[r] *= scale; o1[r] *= scale; }

    // p IS the P^T tile in C-layout. Build B-operand regs:
    // B v0 = P^T[kBase+2*hv][li], v1 = P^T[kBase+2*hv+1][li];
    // element [J][li] lives in p[J%8] at lane li+16*(J/8) -> same lane or xor16.
    float q[8];
#pragma unroll
    for (int r = 0; r < 8; ++r) q[r] = __shfl_xor(p[r], 16, 32);

    v2f bP[4];
    bP[0][0] = hv ? q[2] : p[0];  bP[0][1] = hv ? q[3] : p[1];
    bP[1][0] = hv ? q[6] : p[4];  bP[1][1] = hv ? q[7] : p[5];
    bP[2][0] = hv ? p[2] : q[0];  bP[2][1] = hv ? p[3] : q[1];
    bP[3][0] = hv ? p[6] : q[4];  bP[3][1] = hv ? p[7] : q[5];

    // PV: A = h[c, jBase+..] float2 loads, two 16-row chunks of c
#pragma unroll
    for (int kk = 0; kk < 4; ++kk) {
      v2f aH0 = *(const v2f*)(hb + (size_t)li * N_        + jBase + kk * 4 + 2 * hv);
      v2f aH1 = *(const v2f*)(hb + (size_t)(16 + li) * N_ + jBase + kk * 4 + 2 * hv);
      o0 = WMMA_F32(aH0, bP[kk], o0);
      o1 = WMMA_F32(aH1, bP[kk], o1);
    }
  }

  const float invl = 1.f / l;
  float* ob = o + (size_t)b * HC * N_;
#pragma unroll
  for (int r = 0; r < 8; ++r) {
    ob[(size_t)(r + 8 * hv) * N_      + iBase + li] = o0[r] * invl;
    ob[(size_t)(16 + r + 8 * hv) * N_ + iBase + li] = o1[r] * invl;
  }
}

// ---------------------------------------------------------------------------
// Stage 3: out = lam * (Wv @ o + bv) + x   (memory-bound: 67 MB traffic)
// ---------------------------------------------------------------------------
__global__ __launch_bounds__(256) void out_kernel(
    const float* __restrict__ o, const float* __restrict__ Wv,
    const float* __restrict__ bv, const float* __restrict__ x,
    const float* __restrict__ lam, float* __restrict__ out)
{
  __shared__ float oL[HC][64];
  const int b   = blockIdx.x >> 6;
  const int i0  = (blockIdx.x & 63) * 64;
  const int tid = threadIdx.x;

  const float* ob = o + (size_t)b * HC * N_;
#pragma unroll
  for (int t = 0; t < 8; ++t) {
    int e = tid + t * 256;
    oL[e >> 6][e & 63] = ob[(size_t)(e >> 6) * N_ + i0 + (e & 63)];
  }
  __syncthreads();

  const float lamv = lam[0];
  const int il = tid & 63;
  const int cg = tid >> 6;
  const float* xb = x   + (size_t)b * C_ * N_;
  float*       yb = out + (size_t)b * C_ * N_;

  for (int c = cg; c < C_; c += 4) {
    float acc = bv[c];
    const float* wr = Wv + (size_t)c * HC;
#pragma unroll
    for (int k = 0; k < HC; ++k) acc += wr[k] * oL[k][il];
    size_t idx = (size_t)c * N_ + i0 + il;
    yb[idx] = lamv * acc + xb[idx];
  }
}

// ---------------------------------------------------------------------------
extern "C" void kernel_launch(void* const* d_in, const int* in_sizes, int n_in,
                              void* d_out, int out_size, void* d_ws, size_t ws_size,
                              hipStream_t stream) {
  const float* x   = (const float*)d_in[0];
  const float* Wf  = (const float*)d_in[1];
  const float* bf  = (const float*)d_in[2];
  const float* Wg  = (const float*)d_in[3];
  const float* bg  = (const float*)d_in[4];
  const float* Wh  = (const float*)d_in[5];
  const float* bh  = (const float*)d_in[6];
  const float* Wv  = (const float*)d_in[7];
  const float* bv  = (const float*)d_in[8];
  const float* lam = (const float*)d_in[9];
  float* out = (float*)d_out;

  float* f  = (float*)d_ws;                    // B*FC*N  = 1 MB
  float* gT = f  + (size_t)B_ * FC * N_;       // B*N*FC  = 1 MB
  float* h  = gT + (size_t)B_ * N_ * FC;       // B*HC*N  = 2 MB
  float* o  = h  + (size_t)B_ * HC * N_;       // B*HC*N  = 2 MB

  hipLaunchKernelGGL(fgh_kernel, dim3(B_ * 64), dim3(256), 0, stream,
                     x, Wf, bf, Wg, bg, Wh, bh, f, gT, h);
  hipLaunchKernelGGL(attn_kernel, dim3(B_ * 256), dim3(32), 0, stream,
                     f, gT, h, o);
  hipLaunchKernelGGL(out_kernel, dim3(B_ * 64), dim3(256), 0, stream,
                     o, Wv, bv, x, lam, out);
}